// CompressibleLoss3_61864708931634
// MI455X (gfx1250) — compile-verified
//
#include <hip/hip_runtime.h>

// CompressibleLoss3 for MI455X (gfx1250).
// det(I_p + c A^T A) == det(I_k + c A A^T): shrink 256x256 slogdets to 20x20/10x10.
// One wave32 per pair; gram via V_WMMA_F32_16X16X4_F32; logdet via wave-cooperative
// pivot-free elimination in LDS (matrices are SPD).

#define NUM_AUG    10
#define NPAIRS     500
#define P          256
#define GN         20
#define LDS_STRIDE 260   // floats; bank = 4*row + 2*half + {0,1} -> conflict-free b64

typedef __attribute__((ext_vector_type(2))) float v2f;
typedef __attribute__((ext_vector_type(8))) float v8f;

__device__ __forceinline__ float wave_sum32(float v) {
#pragma unroll
  for (int off = 16; off > 0; off >>= 1)
    v += __shfl_down(v, off, 32);
  return v;  // lane 0 holds the total
}

// In-place no-pivot elimination of SPD n x n matrix in LDS; returns sum(log(diag U))
// in lane 0. Single-wave workgroup: __syncthreads lowers to S_NOP + DS waits.
__device__ float logdet_spd_inplace(float* W, int n, int lane) {
  for (int k = 0; k < n - 1; ++k) {
    const float piv = W[k * n + k];
    if (lane > k && lane < n) {
      const float f = W[lane * n + k] / piv;
      for (int j = k + 1; j < n; ++j)
        W[lane * n + j] -= f * W[k * n + j];
    }
    __syncthreads();
  }
  float v = (lane < n) ? logf(W[lane * n + lane]) : 0.0f;
  return wave_sum32(v);
}

__global__ __launch_bounds__(32)
void pair_gram_logdet(const float* __restrict__ X,
                      const int*   __restrict__ pairs,
                      float* __restrict__ out_lp,
                      float* __restrict__ out_li,
                      float* __restrict__ out_lj) {
  __shared__ float As[32 * LDS_STRIDE];  // 32 padded rows of 256 f32
  __shared__ float G[GN * GN];           // assembled 20x20 gram
  __shared__ float W[GN * GN];           // elimination workspace

  const int s    = blockIdx.x;
  const int lane = threadIdx.x;
  const int pi   = pairs[2 * s + 0];
  const int pj   = pairs[2 * s + 1];

  // ---- stage the 20 selected rows of X into LDS (row t < 10: image i, else image j)
#pragma unroll
  for (int t = 0; t < GN; ++t) {
    const int row = (t < NUM_AUG) ? (pi * NUM_AUG + t) : (pj * NUM_AUG + (t - NUM_AUG));
    const float4* src = (const float4*)(X + (size_t)row * P);
    float4* dst = (float4*)(&As[t * LDS_STRIDE]);
    dst[lane]      = src[lane];        // 64 float4 per row, 2 per lane
    dst[lane + 32] = src[lane + 32];
  }
  {
    const float4 z = make_float4(0.f, 0.f, 0.f, 0.f);
#pragma unroll
    for (int t = GN; t < 32; ++t) {    // zero pad rows so padded tiles are exact
      float4* dst = (float4*)(&As[t * LDS_STRIDE]);
      dst[lane] = z;  dst[lane + 32] = z;
    }
  }
  __syncthreads();

  // ---- G = A A^T via V_WMMA_F32_16X16X4_F32, row blocks {0..15}, {16..31}
  // A 16x4 f32 layout: lane L holds {A[L%16][k0+2*(L/16)], A[..][+1]}; for B = A^T the
  // mirrored layout makes the same register image valid for both operands.
  const int half = lane >> 4;
  const int r    = lane & 15;
  const float* a0p = &As[(r)      * LDS_STRIDE + 2 * half];
  const float* a1p = &As[(16 + r) * LDS_STRIDE + 2 * half];
  v8f c00 = {}; v8f c01 = {}; v8f c11 = {};
#pragma unroll 4
  for (int k0 = 0; k0 < P; k0 += 4) {
    const v2f a0 = *(const v2f*)(a0p + k0);
    const v2f a1 = *(const v2f*)(a1p + k0);
    c00 = __builtin_amdgcn_wmma_f32_16x16x4_f32(false, a0, false, a0, (short)0, c00, false, false);
    c01 = __builtin_amdgcn_wmma_f32_16x16x4_f32(false, a0, false, a1, (short)0, c01, false, false);
    c11 = __builtin_amdgcn_wmma_f32_16x16x4_f32(false, a1, false, a1, (short)0, c11, false, false);
  }

  // ---- scatter C tiles (VGPR v of lane L: M = v + 8*(L/16), N = L%16) into 20x20 G
#pragma unroll
  for (int v = 0; v < 8; ++v) {
    const int m = v + 8 * half;
    G[m * GN + r] = c00[v];                       // tile (0,0): m,n in 0..15
    if (r < 4) {
      G[m * GN + (16 + r)] = c01[v];              // tile (0,1): n in 16..19
      G[(16 + r) * GN + m] = c01[v];              // symmetric mirror (1,0)
    }
    if (half == 0 && v < 4 && r < 4)
      G[(16 + v) * GN + (16 + r)] = c11[v];       // tile (1,1): m,n in 16..19
  }
  __syncthreads();

  const float s20 = 256.0f / ((20.0f + 1e-8f) * 0.01f);  // p / (tr(Pi)+eps)/EPS
  const float s10 = 256.0f / ((10.0f + 1e-8f) * 0.01f);

  // 20x20: I + s20 * G
  if (lane < GN)
    for (int j = 0; j < GN; ++j)
      W[lane * GN + j] = (lane == j ? 1.0f : 0.0f) + s20 * G[lane * GN + j];
  __syncthreads();
  const float lp = logdet_spd_inplace(W, GN, lane);
  __syncthreads();

  // 10x10 image-i block: G[0:10, 0:10]
  if (lane < NUM_AUG)
    for (int j = 0; j < NUM_AUG; ++j)
      W[lane * NUM_AUG + j] = (lane == j ? 1.0f : 0.0f) + s10 * G[lane * GN + j];
  __syncthreads();
  const float li = logdet_spd_inplace(W, NUM_AUG, lane);
  __syncthreads();

  // 10x10 image-j block: G[10:20, 10:20]
  if (lane < NUM_AUG)
    for (int j = 0; j < NUM_AUG; ++j)
      W[lane * NUM_AUG + j] =
          (lane == j ? 1.0f : 0.0f) + s10 * G[(NUM_AUG + lane) * GN + (NUM_AUG + j)];
  __syncthreads();
  const float lj = logdet_spd_inplace(W, NUM_AUG, lane);

  if (lane == 0) { out_lp[s] = lp; out_li[s] = li; out_lj[s] = lj; }
}

__global__ __launch_bounds__(32)
void finalize(const float* __restrict__ lp, const float* __restrict__ li,
              const float* __restrict__ lj, float* __restrict__ out) {
  const int lane = threadIdx.x;
  float sp = 0.f, si = 0.f, sj = 0.f;
  for (int s = lane; s < NPAIRS; s += 32) { sp += lp[s]; si += li[s]; sj += lj[s]; }
  sp = wave_sum32(sp);  si = wave_sum32(si);  sj = wave_sum32(sj);
  if (lane == 0) {
    const float inv = 1.0f / (float)NPAIRS;
    const float mp = sp * inv, mi = si * inv, mj = sj * inv;
    const float ortho = mp - 0.5f * mi - 0.5f * mj;
    out[0] = 0.01f * -ortho;  // GAM3 * -compress_loss_empi_ortho
    out[1] = mp;              // discrimn_loss_empi
    out[2] = mi + mj;         // compress_loss_empi
    out[3] = ortho;           // compress_loss_empi_ortho
  }
}

extern "C" void kernel_launch(void* const* d_in, const int* in_sizes, int n_in,
                              void* d_out, int out_size, void* d_ws, size_t ws_size,
                              hipStream_t stream) {
  (void)in_sizes; (void)n_in; (void)out_size; (void)ws_size;
  const float* X     = (const float*)d_in[0];
  // d_in[1] = y : unused by the loss
  const int*   pairs = (const int*)d_in[2];

  float* wsf = (float*)d_ws;
  float* lp = wsf;          // [500]
  float* li = wsf + 512;    // [500]
  float* lj = wsf + 1024;   // [500]

  pair_gram_logdet<<<NPAIRS, 32, 0, stream>>>(X, pairs, lp, li, lj);
  finalize<<<1, 32, 0, stream>>>(lp, li, lj, (float*)d_out);
}